// HyperConvSingleKernelBlock_83623013253335
// MI455X (gfx1250) — compile-verified
//
#include <hip/hip_runtime.h>

#define B_DIM 128
#define T_DIM 4096
#define CH    64
#define HID   32

#define WS_STRIDE 68   // dwords; (4*lm + k) banks -> conflict-free b64 A-fragment reads
#define XS_STRIDE 40   // dwords; 32-dword rows + 8-dword pad -> disjoint bank groups
#define NT    32       // t-columns per subtile
#define TITER 4        // subtiles per block

typedef __attribute__((ext_vector_type(2))) float v2f;
typedef __attribute__((ext_vector_type(8))) float v8f;

__device__ __forceinline__ float wave_sum32(float v) {
#pragma unroll
  for (int off = 16; off > 0; off >>= 1) v += __shfl_xor(v, off, 32);
  return v;
}

// --- CDNA5 async-to-LDS copy (ASYNCcnt-tracked, bypasses VGPRs) -------------
// global_load_async_to_lds_b128: VDST = VGPR with LDS byte address,
// VADDR = 64-bit global address pair, GV mode (saddr = off).
__device__ __forceinline__ void async_copy_b128(unsigned lds_addr, const float* g) {
  asm volatile("global_load_async_to_lds_b128 %0, %1, off"
               :: "v"(lds_addr), "v"((unsigned long long)(uintptr_t)g)
               : "memory");
}
__device__ __forceinline__ void wait_async_le2() {
  asm volatile("s_wait_asynccnt 0x2" ::: "memory");
}
__device__ __forceinline__ void wait_async_0() {
  asm volatile("s_wait_asynccnt 0x0" ::: "memory");
}
// Generic (flat) pointer to an LDS object: low 32 bits are the LDS byte offset.
__device__ __forceinline__ unsigned lds_addr_of(const void* p) {
  return (unsigned)(uintptr_t)p;
}

// ---------------------------------------------------------------------------
// Hypernetwork: per-sample  Linear(3->32) -> LayerNorm -> ReLU -> Linear(32->N)
// One block per batch sample; wave32 shfl reductions for LayerNorm stats.
// ---------------------------------------------------------------------------
__global__ __launch_bounds__(256)
void hyper_mlp_kernel(const float* __restrict__ z,
                      const float* __restrict__ w_w1, const float* __restrict__ w_b1,
                      const float* __restrict__ w_g,  const float* __restrict__ w_beta,
                      const float* __restrict__ w_w2, const float* __restrict__ w_b2,
                      const float* __restrict__ b_w1, const float* __restrict__ b_b1,
                      const float* __restrict__ b_g,  const float* __restrict__ b_beta,
                      const float* __restrict__ b_w2, const float* __restrict__ b_b2,
                      float* __restrict__ Wout, float* __restrict__ Bout) {
  const int b   = blockIdx.x;
  const int tid = threadIdx.x;
  __shared__ float hW[HID];
  __shared__ float hB[HID];

  const float z0 = z[b * 3 + 0], z1 = z[b * 3 + 1], z2 = z[b * 3 + 2];

  if (tid < 32) {  // wave 0: weight-branch hidden
    float h  = w_w1[tid * 3 + 0] * z0 + w_w1[tid * 3 + 1] * z1 +
               w_w1[tid * 3 + 2] * z2 + w_b1[tid];
    float mu = wave_sum32(h) * (1.0f / 32.0f);
    float d  = h - mu;
    float var = wave_sum32(d * d) * (1.0f / 32.0f);
    float n  = d * rsqrtf(var + 1e-5f) * w_g[tid] + w_beta[tid];
    hW[tid]  = fmaxf(n, 0.0f);
  } else if (tid < 64) {  // wave 1: bias-branch hidden
    const int j = tid - 32;
    float h  = b_w1[j * 3 + 0] * z0 + b_w1[j * 3 + 1] * z1 +
               b_w1[j * 3 + 2] * z2 + b_b1[j];
    float mu = wave_sum32(h) * (1.0f / 32.0f);
    float d  = h - mu;
    float var = wave_sum32(d * d) * (1.0f / 32.0f);
    float n  = d * rsqrtf(var + 1e-5f) * b_g[j] + b_beta[j];
    hB[j]    = fmaxf(n, 0.0f);
  }
  __syncthreads();

  for (int r = tid; r < CH * CH; r += 256) {
    const float* wrow = w_w2 + r * HID;
    float acc = w_b2[r];
#pragma unroll
    for (int k = 0; k < HID; ++k) acc = fmaf(wrow[k], hW[k], acc);
    Wout[(size_t)b * CH * CH + r] = acc;
  }
  if (tid < CH) {
    const float* brow = b_w2 + tid * HID;
    float acc = b_b2[tid];
#pragma unroll
    for (int k = 0; k < HID; ++k) acc = fmaf(brow[k], hB[k], acc);
    Bout[b * CH + tid] = acc;
  }
}

// ---------------------------------------------------------------------------
// Batched 1x1 conv as WMMA f32 GEMM:  out[b] = W[b](64x64) @ x[b](64x4096) + bias
// Grid (T/128, B); block = 256 threads = 8 wave32.
// x tiles double-buffered in LDS via global_load_async_to_lds_b128 (ASYNCcnt),
// A fragments register-resident, K=64 via 16x V_WMMA_F32_16X16X4_F32.
// ---------------------------------------------------------------------------
__global__ __launch_bounds__(256)
void hyperconv_wmma_kernel(const float* __restrict__ X, const float* __restrict__ W,
                           const float* __restrict__ Bias, float* __restrict__ Out) {
  const int b      = blockIdx.y;
  const int tblock = blockIdx.x * (NT * TITER);
  const int tid    = threadIdx.x;
  const int lane   = tid & 31;
  const int wave   = tid >> 5;
  const int m0     = (wave & 3) << 4;
  const int n0     = (wave >> 2) << 4;
  const int lm     = lane & 15;
  const int half   = lane >> 4;

  __shared__ float ws[CH * WS_STRIDE];          // staged W[b], padded
  __shared__ float xs[2][CH * XS_STRIDE];       // double-buffered x tiles, padded

  // Stage W[b] (16 KB) into LDS with coalesced float4 loads (L2-resident).
  const float* Wb = W + (size_t)b * CH * CH;
  for (int i = tid; i < CH * CH / 4; i += 256) {
    const int r = i >> 4;
    const int c = (i & 15) << 2;
    const float4 v = *(const float4*)(Wb + r * CH + c);
    float* d = &ws[r * WS_STRIDE + c];
    d[0] = v.x; d[1] = v.y; d[2] = v.z; d[3] = v.w;
  }

  const float* Xb = X + (size_t)b * CH * T_DIM;

  // Per-thread async copy slots: 2x b128 per subtile (64 rows x 8 chunks).
  const int r0 = tid >> 3;            // rows 0..31   (i = tid)
  const int r1 = (tid + 256) >> 3;    // rows 32..63  (i = tid + 256)
  const int c0 = (tid & 7) << 2;      // chunk start column (floats)
  const unsigned lds0a = lds_addr_of(&xs[0][r0 * XS_STRIDE + c0]);
  const unsigned lds0b = lds_addr_of(&xs[0][r1 * XS_STRIDE + c0]);
  const unsigned lds_buf_stride = (unsigned)(CH * XS_STRIDE * sizeof(float));
  const float* g0 = Xb + (size_t)r0 * T_DIM + c0;
  const float* g1 = Xb + (size_t)r1 * T_DIM + c0;

  // Prologue: kick off subtile 0 into buffer 0.
  async_copy_b128(lds0a, g0 + tblock);
  async_copy_b128(lds0b, g1 + tblock);

  __syncthreads();  // W staged (also orders LDS init vs. A-fragment reads)

  // A fragments: 16x4 f32 per k-step. lanes 0-15 hold K=k0,k0+1; lanes 16-31
  // hold K=k0+2,k0+3 for row M = m0 + (lane&15).
  v2f a[16];
  {
    const float* ap = &ws[(m0 + lm) * WS_STRIDE + 2 * half];
#pragma unroll
    for (int ks = 0; ks < 16; ++ks) a[ks] = *(const v2f*)(ap + ks * 4);
  }

  // Bias for this wave's 8 output rows (C/D layout: vgpr v -> row m0 + v + 8*half).
  float bias8[8];
  {
    const float* bp = Bias + b * CH + m0 + 8 * half;
#pragma unroll
    for (int v = 0; v < 8; ++v) bias8[v] = bp[v];
  }

  float* Ob = Out + (size_t)b * CH * T_DIM + (size_t)(m0 + 8 * half) * T_DIM;

  for (int it = 0; it < TITER; ++it) {
    const int cur = it & 1;
    const int t0  = tblock + it * NT;

    // Kick off the next subtile into the other buffer (its consumers finished
    // before the end-of-iteration barrier of it-1).
    if (it + 1 < TITER) {
      const unsigned nbuf = (unsigned)(1 - cur) * lds_buf_stride;
      async_copy_b128(lds0a + nbuf, g0 + t0 + NT);
      async_copy_b128(lds0b + nbuf, g1 + t0 + NT);
      wait_async_le2();   // in-order retirement: current buffer's loads landed
    } else {
      wait_async_0();
    }
    __syncthreads();      // all waves' async writes for buf[cur] visible

    v8f acc = {};
    // B fragments: 4x16 f32. lanes 0-15 hold rows k0,k0+1; lanes 16-31 rows
    // k0+2,k0+3, column N = n0 + (lane&15).
    const float* bb = &xs[cur][(2 * half) * XS_STRIDE + n0 + lm];
#pragma unroll
    for (int ks = 0; ks < 16; ++ks) {
      v2f bf;
      bf.x = bb[ks * 4 * XS_STRIDE];
      bf.y = bb[ks * 4 * XS_STRIDE + XS_STRIDE];
      acc = __builtin_amdgcn_wmma_f32_16x16x4_f32(
          /*neg_a=*/false, a[ks], /*neg_b=*/false, bf,
          /*c_mod=*/(short)0, acc, /*reuse_a=*/false, /*reuse_b=*/false);
    }

    // Store with fused bias, non-temporal (streamed output, never re-read).
    float* op = Ob + t0 + n0 + lm;
#pragma unroll
    for (int v = 0; v < 8; ++v)
      __builtin_nontemporal_store(acc[v] + bias8[v], op + (size_t)v * T_DIM);

    __syncthreads();      // buf[cur] consumed by all waves before reissue
  }
}

extern "C" void kernel_launch(void* const* d_in, const int* in_sizes, int n_in,
                              void* d_out, int out_size, void* d_ws, size_t ws_size,
                              hipStream_t stream) {
  const float* x      = (const float*)d_in[0];
  const float* z      = (const float*)d_in[1];
  const float* w_w1   = (const float*)d_in[2];
  const float* w_b1   = (const float*)d_in[3];
  const float* w_g    = (const float*)d_in[4];
  const float* w_beta = (const float*)d_in[5];
  const float* w_w2   = (const float*)d_in[6];
  const float* w_b2   = (const float*)d_in[7];
  const float* b_w1   = (const float*)d_in[8];
  const float* b_b1   = (const float*)d_in[9];
  const float* b_g    = (const float*)d_in[10];
  const float* b_beta = (const float*)d_in[11];
  const float* b_w2   = (const float*)d_in[12];
  const float* b_b2   = (const float*)d_in[13];

  float* Wgen = (float*)d_ws;                       // [128, 64, 64]
  float* Bgen = Wgen + (size_t)B_DIM * CH * CH;     // [128, 64]

  hyper_mlp_kernel<<<B_DIM, 256, 0, stream>>>(
      z, w_w1, w_b1, w_g, w_beta, w_w2, w_b2,
      b_w1, b_b1, b_g, b_beta, b_w2, b_b2, Wgen, Bgen);

  dim3 grid(T_DIM / (NT * TITER), B_DIM);
  hyperconv_wmma_kernel<<<grid, 256, 0, stream>>>(x, Wgen, Bgen, (float*)d_out);
}